// model_GCN_54769422958882
// MI455X (gfx1250) — compile-verified
//
#include <hip/hip_runtime.h>
#include <hip/hip_bf16.h>
#include <math.h>

typedef __attribute__((ext_vector_type(2))) float v2f;
typedef __attribute__((ext_vector_type(8))) float v8f;

#define NNODES 100000
#define NEDGES 1600000
#define HFEAT  32
#define GRAPHS 64
#define COUT   8

// ---------------------------------------------------------------------------
// Normalization precompute
// ---------------------------------------------------------------------------
__global__ __launch_bounds__(256) void deg_init_kernel(float* __restrict__ deg) {
    int i = blockIdx.x * blockDim.x + threadIdx.x;
    if (i < NNODES) deg[i] = 1.0f;   // self-loop weight folded in
}

__global__ __launch_bounds__(256) void deg_accum_kernel(const int* __restrict__ eidx,
                                                        const float* __restrict__ w,
                                                        float* __restrict__ deg) {
    int e = blockIdx.x * blockDim.x + threadIdx.x;
    if (e < NEDGES) atomicAdd(&deg[eidx[NEDGES + e]], w[e]);  // target = col
}

__global__ __launch_bounds__(256) void deg_to_dinv_kernel(float* __restrict__ deg) {
    int i = blockIdx.x * blockDim.x + threadIdx.x;
    if (i < NNODES) {
        float d = deg[i];
        deg[i] = (d > 0.0f) ? rsqrtf(d) : 0.0f;
    }
}

__global__ __launch_bounds__(256) void edge_norm_kernel(const int* __restrict__ eidx,
                                                        const float* __restrict__ w,
                                                        const float* __restrict__ dinv,
                                                        float* __restrict__ norm) {
    int e = blockIdx.x * blockDim.x + threadIdx.x;
    if (e < NEDGES) norm[e] = dinv[eidx[e]] * w[e] * dinv[eidx[NEDGES + e]];
}

// ---------------------------------------------------------------------------
// Dense projection  Y[N,32] = (relu?)X[N,32] @ W[32,32]   via V_WMMA_F32_16X16X4_F32
// One wave per 16-row tile; two 16x16 accumulators cover the 32 output cols.
// A(16x4) layout: lanes 0-15 hold M=lane, K = k0+{0,1}; lanes 16-31 K = k0+{2,3}.
// B(4x16) layout mirrors A (col = lane&15).  C/D: VGPR r = row r (lanes 0-15)
// or row r+8 (lanes 16-31), col = lane&15.
// ---------------------------------------------------------------------------
__global__ __launch_bounds__(256) void gcn_gemm_wmma_kernel(const float* __restrict__ X,
                                                            const float* __restrict__ W,
                                                            float* __restrict__ Y,
                                                            int nrows, int relu_in) {
    int wave = (int)((blockIdx.x * blockDim.x + threadIdx.x) >> 5);
    int lane = threadIdx.x & 31;
    int row0 = wave * 16;
    if (row0 >= nrows) return;                 // wave-uniform; EXEC stays all-1s

    int m     = lane & 15;
    int khalf = (lane >> 4) << 1;              // 0 for lanes 0-15, 2 for 16-31

    v8f acc0 = {};
    v8f acc1 = {};
    const float* xrow = X + (size_t)(row0 + m) * HFEAT;

#pragma unroll
    for (int k0 = 0; k0 < HFEAT; k0 += 4) {
        int kb = k0 + khalf;
        float ax = xrow[kb];
        float ay = xrow[kb + 1];
        if (relu_in) { ax = fmaxf(ax, 0.0f); ay = fmaxf(ay, 0.0f); }
        v2f a;  a.x = ax;                 a.y = ay;
        v2f b0; b0.x = W[kb * 32 + m];        b0.y = W[(kb + 1) * 32 + m];
        v2f b1; b1.x = W[kb * 32 + 16 + m];   b1.y = W[(kb + 1) * 32 + 16 + m];

        acc0 = __builtin_amdgcn_wmma_f32_16x16x4_f32(false, a, false, b0,
                                                     (short)0, acc0, false, false);
        acc1 = __builtin_amdgcn_wmma_f32_16x16x4_f32(false, a, false, b1,
                                                     (short)0, acc1, false, false);
    }

    int rbase = row0 + ((lane >> 4) << 3);     // +8 rows for upper half-wave
#pragma unroll
    for (int r = 0; r < 8; ++r) {
        Y[(size_t)(rbase + r) * HFEAT + m]      = acc0[r];
        Y[(size_t)(rbase + r) * HFEAT + 16 + m] = acc1[r];
    }
}

// ---------------------------------------------------------------------------
// agg = bias + h * dinv^2   (self-loop contribution, norm_ii = dinv_i * 1 * dinv_i)
// ---------------------------------------------------------------------------
__global__ __launch_bounds__(256) void agg_init_kernel(const float* __restrict__ H,
                                                       const float* __restrict__ dinv,
                                                       const float* __restrict__ bias,
                                                       float* __restrict__ agg) {
    int i = blockIdx.x * blockDim.x + threadIdx.x;
    if (i >= NNODES * HFEAT) return;
    int node = i >> 5;
    int f    = i & 31;
    float di = dinv[node];
    agg[i] = bias[f] + H[i] * di * di;
}

// ---------------------------------------------------------------------------
// Message gather + scatter-add: one wave per edge, lane = feature.
// 128 B coalesced load + 128 B coalesced global fp32 atomic per edge.
// ---------------------------------------------------------------------------
__global__ __launch_bounds__(256) void edge_scatter_kernel(const float* __restrict__ H,
                                                           const float* __restrict__ norm,
                                                           const int* __restrict__ eidx,
                                                           float* __restrict__ agg) {
    int wave = (int)((blockIdx.x * blockDim.x + threadIdx.x) >> 5);
    int lane = threadIdx.x & 31;
    if (wave >= NEDGES) return;
    int   r   = eidx[wave];            // source node
    int   c   = eidx[NEDGES + wave];   // target node
    float nrm = norm[wave];
    float v = H[(size_t)r * HFEAT + lane] * nrm;
    atomicAdd(&agg[(size_t)c * HFEAT + lane], v);
}

// ---------------------------------------------------------------------------
// Global mean pooling (with relu applied to final layer output)
// ---------------------------------------------------------------------------
__global__ __launch_bounds__(256) void pool_init_kernel(float* __restrict__ sums,
                                                        float* __restrict__ counts) {
    int i = blockIdx.x * blockDim.x + threadIdx.x;
    if (i < GRAPHS * HFEAT) sums[i] = 0.0f;
    if (i < GRAPHS) counts[i] = 0.0f;
}

__global__ __launch_bounds__(256) void pool_accum_kernel(const float* __restrict__ H,
                                                         const int* __restrict__ batch,
                                                         float* __restrict__ sums,
                                                         float* __restrict__ counts) {
    int wave = (int)((blockIdx.x * blockDim.x + threadIdx.x) >> 5);
    int lane = threadIdx.x & 31;
    if (wave >= NNODES) return;
    int g = batch[wave];
    float v = fmaxf(H[(size_t)wave * HFEAT + lane], 0.0f);
    atomicAdd(&sums[g * HFEAT + lane], v);
    if (lane == 0) atomicAdd(&counts[g], 1.0f);
}

// ---------------------------------------------------------------------------
// Head: pooled @ Wout + bout, then log_softmax.  64 graphs -> one thread each.
// ---------------------------------------------------------------------------
__global__ __launch_bounds__(64) void head_kernel(const float* __restrict__ sums,
                                                  const float* __restrict__ counts,
                                                  const float* __restrict__ Wout,
                                                  const float* __restrict__ bout,
                                                  float* __restrict__ out) {
    int g = threadIdx.x;
    if (g >= GRAPHS) return;
    float inv = 1.0f / fmaxf(counts[g], 1.0f);
    float logits[COUT];
#pragma unroll
    for (int c = 0; c < COUT; ++c) logits[c] = bout[c];
    for (int f = 0; f < HFEAT; ++f) {
        float p = sums[g * HFEAT + f] * inv;
#pragma unroll
        for (int c = 0; c < COUT; ++c) logits[c] += p * Wout[f * COUT + c];
    }
    float m = logits[0];
#pragma unroll
    for (int c = 1; c < COUT; ++c) m = fmaxf(m, logits[c]);
    float s = 0.0f;
#pragma unroll
    for (int c = 0; c < COUT; ++c) s += expf(logits[c] - m);
    float lse = m + logf(s);
#pragma unroll
    for (int c = 0; c < COUT; ++c) out[g * COUT + c] = logits[c] - lse;
}

// ---------------------------------------------------------------------------
// Launch
// ---------------------------------------------------------------------------
extern "C" void kernel_launch(void* const* d_in, const int* in_sizes, int n_in,
                              void* d_out, int out_size, void* d_ws, size_t ws_size,
                              hipStream_t stream) {
    (void)in_sizes; (void)n_in; (void)out_size; (void)ws_size;

    const float* x    = (const float*)d_in[0];
    const int*   eidx = (const int*)  d_in[1];
    const float* ew   = (const float*)d_in[2];
    const int*   bat  = (const int*)  d_in[3];
    const float* W1   = (const float*)d_in[4];
    const float* b1   = (const float*)d_in[5];
    const float* W2   = (const float*)d_in[6];
    const float* b2   = (const float*)d_in[7];
    const float* W3   = (const float*)d_in[8];
    const float* b3   = (const float*)d_in[9];
    const float* Wout = (const float*)d_in[10];
    const float* bout = (const float*)d_in[11];
    float* out = (float*)d_out;

    // Workspace carve-up (all offsets 256B-aligned)
    char* ws = (char*)d_ws;
    size_t off = 0;
    auto carve = [&](size_t bytes) { char* p = ws + off; off += (bytes + 255) & ~(size_t)255; return p; };
    float* dinv   = (float*)carve((size_t)NNODES * sizeof(float));
    float* norm   = (float*)carve((size_t)NEDGES * sizeof(float));
    float* hA     = (float*)carve((size_t)NNODES * HFEAT * sizeof(float));
    float* hB     = (float*)carve((size_t)NNODES * HFEAT * sizeof(float));
    float* sums   = (float*)carve((size_t)GRAPHS * HFEAT * sizeof(float));
    float* counts = (float*)carve((size_t)GRAPHS * sizeof(float));

    const int TB = 256;
    int gNode   = (NNODES + TB - 1) / TB;                 // thread per node
    int gEdge   = (NEDGES + TB - 1) / TB;                 // thread per edge
    int gNodeF  = (NNODES * HFEAT + TB - 1) / TB;         // thread per (node,feat)
    int gEdgeW  = (NEDGES * 32 + TB - 1) / TB;            // wave per edge
    int gNodeW  = (NNODES * 32 + TB - 1) / TB;            // wave per node
    int nTiles  = (NNODES + 15) / 16;
    int gGemm   = (nTiles * 32 + TB - 1) / TB;            // wave per 16-row tile

    // --- normalization ---
    deg_init_kernel<<<gNode, TB, 0, stream>>>(dinv);
    deg_accum_kernel<<<gEdge, TB, 0, stream>>>(eidx, ew, dinv);
    deg_to_dinv_kernel<<<gNode, TB, 0, stream>>>(dinv);
    edge_norm_kernel<<<gEdge, TB, 0, stream>>>(eidx, ew, dinv, norm);

    // --- layer 1 ---
    gcn_gemm_wmma_kernel<<<gGemm, TB, 0, stream>>>(x, W1, hA, NNODES, 0);
    agg_init_kernel<<<gNodeF, TB, 0, stream>>>(hA, dinv, b1, hB);
    edge_scatter_kernel<<<gEdgeW, TB, 0, stream>>>(hA, norm, eidx, hB);

    // --- layer 2 (relu fused into GEMM input) ---
    gcn_gemm_wmma_kernel<<<gGemm, TB, 0, stream>>>(hB, W2, hA, NNODES, 1);
    agg_init_kernel<<<gNodeF, TB, 0, stream>>>(hA, dinv, b2, hB);
    edge_scatter_kernel<<<gEdgeW, TB, 0, stream>>>(hA, norm, eidx, hB);

    // --- layer 3 ---
    gcn_gemm_wmma_kernel<<<gGemm, TB, 0, stream>>>(hB, W3, hA, NNODES, 1);
    agg_init_kernel<<<gNodeF, TB, 0, stream>>>(hA, dinv, b3, hB);
    edge_scatter_kernel<<<gEdgeW, TB, 0, stream>>>(hA, norm, eidx, hB);

    // --- pooling + head (relu fused into pooling) ---
    pool_init_kernel<<<(GRAPHS * HFEAT + TB - 1) / TB, TB, 0, stream>>>(sums, counts);
    pool_accum_kernel<<<gNodeW, TB, 0, stream>>>(hB, bat, sums, counts);
    head_kernel<<<1, 64, 0, stream>>>(sums, counts, Wout, bout, out);
}